// CornLoss_17626545783550
// MI455X (gfx1250) — compile-verified
//
#include <hip/hip_runtime.h>

// ---------------------------------------------------------------------------
// CORN ordinal loss, MI455X (gfx1250, wave32).
// Memory-bound: ~170 MB streamed, ~7.5 us at 23.3 TB/s HBM. Strategy:
//   - 128-bit coalesced non-temporal loads of logits (single-use stream)
//   - global_prefetch_b8 one grid-stride tile ahead
//   - 1 exp + 1 rcp + 1 log per element (trans units, far under roofline)
//   - deterministic two-kernel reduction (no float atomics -> graph-stable)
//   - final 32-lane reduction via V_WMMA_F32_16X16X4_F32 (ones-vector matmul)
// ---------------------------------------------------------------------------

#define CORN_EPS 1e-7f

typedef float v8f __attribute__((ext_vector_type(8)));
typedef float v4f __attribute__((ext_vector_type(4)));
typedef float v2f __attribute__((ext_vector_type(2)));

__device__ __forceinline__ float corn_elem(float x, unsigned col, int rank0) {
    // p = sigmoid(x), q = 1 - sigmoid(x), computed with one exp + one rcp
    float e = __expf(-x);
    float r = __builtin_amdgcn_rcpf(1.0f + e);
    float p = r;
    float q = e * r;
    float v = ((int)col < rank0) ? (p + CORN_EPS) : (q + CORN_EPS);
    return -__logf(v);
}

__device__ __forceinline__ float wave32_reduce(float v) {
    #pragma unroll
    for (int off = 16; off > 0; off >>= 1)
        v += __shfl_xor(v, off, 32);
    return v;
}

// ---------------------------------------------------------------------------
// Kernel 1: streaming partial sums, one float per block into d_ws.
// ---------------------------------------------------------------------------
__global__ __launch_bounds__(256) void corn_partial_kernel(
    const float* __restrict__ logits,
    const int* __restrict__ targets,
    float* __restrict__ partials,
    unsigned n4, unsigned n)
{
    const unsigned tid    = blockIdx.x * blockDim.x + threadIdx.x;
    const unsigned stride = gridDim.x * blockDim.x;
    const v4f* l4 = (const v4f*)logits;

    float acc = 0.0f;
    for (unsigned i = tid; i < n4; i += stride) {
        // Prefetch one grid-stride tile ahead (speculative; OOB is dropped).
        __builtin_prefetch(&l4[i + stride], 0, 0);
        v4f x = __builtin_nontemporal_load(&l4[i]);   // global_load_b128, NT
        unsigned base = i * 4u;
        #pragma unroll
        for (int j = 0; j < 4; ++j) {
            unsigned idx = base + (unsigned)j;
            unsigned row = idx / 9u;                  // mulhi via compiler magic
            unsigned col = idx - row * 9u;
            int rank0 = targets[row] - 1;             // broadcast within wave, L0/L2 hit
            acc += corn_elem(x[j], col, rank0);
        }
    }
    // Scalar tail (n not divisible by 4 — not hit for this shape, kept general).
    for (unsigned idx = n4 * 4u + tid; idx < n; idx += stride) {
        unsigned row = idx / 9u;
        unsigned col = idx - row * 9u;
        acc += corn_elem(logits[idx], col, targets[row] - 1);
    }

    // wave32 -> LDS -> block partial
    acc = wave32_reduce(acc);
    __shared__ float smem[8];
    const unsigned lane = threadIdx.x & 31u;
    const unsigned wave = threadIdx.x >> 5;
    if (lane == 0) smem[wave] = acc;
    __syncthreads();
    if (wave == 0) {
        float s = (lane < (blockDim.x >> 5)) ? smem[lane] : 0.0f;
        s = wave32_reduce(s);
        if (lane == 0) partials[blockIdx.x] = s;
    }
}

// ---------------------------------------------------------------------------
// Kernel 2: single-block final reduction. The last 32->1 stage uses
// V_WMMA_F32_16X16X4_F32 with an all-ones B matrix: D[m,n] = sum_k A[m,k],
// so packing the 32 lane partials into A (a0=val, a1=0) and summing D's 8
// accumulator VGPRs on lanes 0/16 (+ one shfl_xor 16) gives the exact-f32
// total of all 32 lanes.
// ---------------------------------------------------------------------------
__global__ __launch_bounds__(256) void corn_final_kernel(
    const float* __restrict__ partials,
    float* __restrict__ out,
    unsigned n, float inv_n)
{
    float acc = 0.0f;
    for (unsigned i = threadIdx.x; i < n; i += blockDim.x)
        acc += partials[i];

    acc = wave32_reduce(acc);
    __shared__ float smem[8];
    const unsigned lane = threadIdx.x & 31u;
    const unsigned wave = threadIdx.x >> 5;
    if (lane == 0) smem[wave] = acc;
    __syncthreads();

    if (wave == 0) {                         // whole wave enters: EXEC all-1s
        float s = (lane < 8) ? smem[lane] : 0.0f;
#if __has_builtin(__builtin_amdgcn_wmma_f32_16x16x4_f32)
        v2f a; a.x = s;    a.y = 0.0f;       // A 16x4 f32: 2 VGPRs/lane
        v2f b; b.x = 1.0f; b.y = 1.0f;       // B 4x16 f32: all ones
        v8f c = {};
        v8f d = __builtin_amdgcn_wmma_f32_16x16x4_f32(
            /*neg_a=*/false, a, /*neg_b=*/false, b,
            /*c_mod=*/(short)0, c, /*reuse_a=*/false, /*reuse_b=*/false);
        // lane0 holds row-sums M=0..7 in d[0..7]; lane16 holds M=8..15.
        float t = d[0] + d[1] + d[2] + d[3] + d[4] + d[5] + d[6] + d[7];
        t += __shfl_xor(t, 16, 32);
#else
        float t = wave32_reduce(s);
#endif
        if (lane == 0) out[0] = t * inv_n;
    }
}

// ---------------------------------------------------------------------------
extern "C" void kernel_launch(void* const* d_in, const int* in_sizes, int n_in,
                              void* d_out, int out_size, void* d_ws, size_t ws_size,
                              hipStream_t stream) {
    const float* logits  = (const float*)d_in[0];
    const int*   targets = (const int*)d_in[1];
    float*       out     = (float*)d_out;
    float*       partials = (float*)d_ws;

    const unsigned n  = (unsigned)in_sizes[0];   // 37,748,736 loss elements
    const unsigned n4 = n / 4u;

    const int TPB    = 256;                      // 8 wave32 waves per block
    const int BLOCKS = 2048;                     // partials fit easily in d_ws

    corn_partial_kernel<<<BLOCKS, TPB, 0, stream>>>(logits, targets, partials, n4, n);
    corn_final_kernel<<<1, TPB, 0, stream>>>(partials, out, (unsigned)BLOCKS,
                                             1.0f / (float)n);
}